// MatchNet_1228360646760
// MI455X (gfx1250) — compile-verified
//
#include <hip/hip_runtime.h>
#include <hip/hip_bf16.h>
#include <math.h>

#define N_ITERS 600
#define CONTROL 10.0f

typedef float v2f __attribute__((ext_vector_type(2)));
typedef float v8f __attribute__((ext_vector_type(8)));

// ---------------------------------------------------------------------------
// Kernel 1: 4-layer tanh MLP.  One block per batch row.
// ---------------------------------------------------------------------------
__global__ __launch_bounds__(128)
void mlp_kernel(const float* __restrict__ X,
                const float* __restrict__ W1, const float* __restrict__ b1,
                const float* __restrict__ W2, const float* __restrict__ b2,
                const float* __restrict__ W3, const float* __restrict__ b3,
                const float* __restrict__ W4, const float* __restrict__ b4,
                float* __restrict__ z)
{
    __shared__ float xr[64];
    __shared__ float h1[20], h2[20], h3[20];
    const int tid = threadIdx.x;
    const int row = blockIdx.x;

    if (tid < 64) xr[tid] = X[row * 64 + tid];
    __syncthreads();

    if (tid < 20) {
        float s = b1[tid];
        for (int k = 0; k < 64; ++k) s += xr[k] * W1[k * 20 + tid];
        h1[tid] = tanhf(s);
    }
    __syncthreads();
    if (tid < 20) {
        float s = b2[tid];
        for (int k = 0; k < 20; ++k) s += h1[k] * W2[k * 20 + tid];
        h2[tid] = tanhf(s);
    }
    __syncthreads();
    if (tid < 20) {
        float s = b3[tid];
        for (int k = 0; k < 20; ++k) s += h2[k] * W3[k * 20 + tid];
        h3[tid] = tanhf(s);
    }
    __syncthreads();
    for (int n = tid; n < 512; n += 128) {
        float s = b4[n];
        for (int k = 0; k < 20; ++k) s += h3[k] * W4[k * 512 + n];
        z[row * 512 + n] = s;
    }
}

// ---------------------------------------------------------------------------
// Kernel 2: power iteration for the primal/dual step size.
// ---------------------------------------------------------------------------
__global__ __launch_bounds__(512)
void power_kernel(const float* __restrict__ S, float* __restrict__ stepOut)
{
    __shared__ float v[512];
    __shared__ float u[64];
    __shared__ float red;
    const int tid = threadIdx.x;

    v[tid] = 1.0f;
    __syncthreads();

    for (int it = 0; it < 30; ++it) {
        if (tid < 64) {
            float s = 0.0f;
            for (int j = 0; j < 512; ++j) s += S[tid * 512 + j] * v[j];
            u[tid] = s;
        }
        if (tid == 0) red = 0.0f;
        __syncthreads();
        float s = 0.0f;
        for (int i = 0; i < 64; ++i) s += S[i * 512 + tid] * u[i];
        atomicAdd(&red, s * s);
        __syncthreads();
        const float nrm = sqrtf(red);
        v[tid] = s / (nrm + 1e-12f);
        __syncthreads();
    }
    if (tid == 0) red = 0.0f;
    __syncthreads();
    if (tid < 64) {
        float s = 0.0f;
        for (int j = 0; j < 512; ++j) s += S[tid * 512 + j] * v[j];
        atomicAdd(&red, s * s);
    }
    __syncthreads();
    if (tid == 0) stepOut[0] = 0.99f / sqrtf(red + 1.0f);
}

// ---------------------------------------------------------------------------
// Kernel 3: 600-iteration Chambolle-Pock solve.
// Block = 16 batch rows, 8 waves; wave w owns n-columns [64w, 64w+64).
// x / y2 / b live in registers; z in LDS (read 1x/iter); xb does an LDS
// round-trip (lane relayout for GEMM2-A).  S is pre-packed at init into
// fragment-ordered u16 streams so each lane pulls 8 K-steps of B-data per
// ds_load_b128, unpacked with v_cvt_f32_ubyte0..3 only.
// 3 block barriers per iteration.
// ---------------------------------------------------------------------------
__global__ __launch_bounds__(256)
void solver_kernel(const float* __restrict__ X,    // [64][64]  constraint rhs b
                   const float* __restrict__ Sg,   // [64][512] 0/1 matrix
                   const float* __restrict__ zg,   // [64][512] MLP output
                   const float* __restrict__ stepp,
                   float* __restrict__ out)        // [64][512]
{
    __shared__ float xbs[16][512];                        // xb relayout buffer
    __shared__ float zs [16][512];
    __shared__ float y1s[16][64];
    __shared__ __align__(16) float y1acc[16][64];
    // Fragment-packed S: u16 = byte pair (k+2lh, k+2lh+1) for one K-step.
    __shared__ __align__(16) unsigned short Stp[512][2][16];  // GEMM1-B: [n][lh][ks]
    __shared__ __align__(16) unsigned short Ssp[64][2][128];  // GEMM2-B: [m][lh][kk]
    __shared__ float pnorm[16];

    const int tid  = threadIdx.x;
    const int wave = tid >> 5;
    const int lane = tid & 31;
    const int lh   = lane >> 4;       // lane half
    const int l16  = lane & 15;
    const int r0   = blockIdx.x * 16; // batch-row base
    const int c0   = wave * 64;       // wave's n-column base
    const int kb   = wave * 16;       // wave's K-step base for GEMM2

    const float stepv = stepp[0];
    const float tau = stepv, sigma = stepv;

    __builtin_prefetch(Sg + tid * 64, 0, 3);
    __builtin_prefetch(zg + (r0 + (tid >> 4)) * 512, 0, 3);

    // ---- one-time LDS setup: fragment-packed S + z ----
    for (int i = tid; i < 512 * 2 * 16; i += 256) {       // Stp[n][lh][ks]
        const int n = i >> 5, rem = i & 31;
        const int lhh = rem >> 4, ks = rem & 15;
        const unsigned int b0 = (Sg[(4 * ks + 2 * lhh) * 512 + n] != 0.0f);
        const unsigned int b1 = (Sg[(4 * ks + 2 * lhh + 1) * 512 + n] != 0.0f);
        ((unsigned short*)Stp)[i] = (unsigned short)(b0 | (b1 << 8));
    }
    for (int i = tid; i < 64 * 2 * 128; i += 256) {       // Ssp[m][lh][kk]
        const int nn = i >> 8, rem = i & 255;
        const int lhh = rem >> 7, kk = rem & 127;
        const unsigned int b0 = (Sg[nn * 512 + 4 * kk + 2 * lhh] != 0.0f);
        const unsigned int b1 = (Sg[nn * 512 + 4 * kk + 2 * lhh + 1] != 0.0f);
        ((unsigned short*)Ssp)[i] = (unsigned short)(b0 | (b1 << 8));
    }
    for (int i = tid; i < 16 * 512; i += 256) {
        const int r = i >> 9, c = i & 511;
        zs[r][c] = zg[(r0 + r) * 512 + c];
    }
    float br[4];                                          // b at this thread's y1 slots
    #pragma unroll
    for (int q = 0; q < 4; ++q) {
        const int idx = tid + q * 256;
        const int r = idx >> 6, c = idx & 63;
        br[q] = X[(r0 + r) * 64 + c];
        y1s[r][c] = 0.0f;
    }
    __syncthreads();

    // ---- wave-private register state (x, y2) ----
    v8f xr[4], y2r[4];
    #pragma unroll
    for (int t = 0; t < 4; ++t) {
        const int n = c0 + t * 16 + l16;
        #pragma unroll
        for (int j = 0; j < 8; ++j) {
            xr[t][j]  = fmaxf(zs[j + 8 * lh][n], 0.0f);
            y2r[t][j] = 0.0f;
        }
    }

    for (int it = 0; it < N_ITERS; ++it) {
        // ---- zero accumulators
        if (tid < 16) pnorm[tid] = 0.0f;
        ((float4*)y1acc)[tid] = make_float4(0.f, 0.f, 0.f, 0.f);
        __syncthreads();

        // ---- GEMM1: t1 = y1 @ S (own columns).  A-frags hoisted (K-only).
        v2f a1[16];
        #pragma unroll
        for (int ks = 0; ks < 16; ++ks) {
            a1[ks].x = y1s[l16][4 * ks + 2 * lh];
            a1[ks].y = y1s[l16][4 * ks + 2 * lh + 1];
        }
        float sums[8] = {0.f,0.f,0.f,0.f,0.f,0.f,0.f,0.f};
        #pragma unroll
        for (int t = 0; t < 4; ++t) {
            v8f acc = {0.f,0.f,0.f,0.f,0.f,0.f,0.f,0.f};
            const int n = c0 + t * 16 + l16;
            const uint4 Wa = *(const uint4*)&Stp[n][lh][0];   // ks 0..7
            const uint4 Wb = *(const uint4*)&Stp[n][lh][8];   // ks 8..15
            const unsigned int wbuf[8] = {Wa.x, Wa.y, Wa.z, Wa.w,
                                          Wb.x, Wb.y, Wb.z, Wb.w};
            #pragma unroll
            for (int ks = 0; ks < 16; ++ks) {
                const unsigned int w = wbuf[ks >> 1];
                v2f b;
                if ((ks & 1) == 0) { b.x = (float)(w & 0xffu);
                                     b.y = (float)((w >> 8) & 0xffu); }
                else               { b.x = (float)((w >> 16) & 0xffu);
                                     b.y = (float)(w >> 24); }
                acc = __builtin_amdgcn_wmma_f32_16x16x4_f32(
                        false, a1[ks], false, b, (short)0, acc, false, false);
            }
            // primal update + dual-2 pre-projection
            #pragma unroll
            for (int j = 0; j < 8; ++j) {
                const int r = j + 8 * lh;
                const float xo  = xr[t][j];
                const float xnv = fmaxf(xo - tau * (acc[j] + y2r[t][j]) + tau, 0.0f);
                const float xbv = 2.0f * xnv - xo;
                xr[t][j] = xnv;
                xbs[r][n] = xbv;                              // relayout for GEMM2-A
                const float pv = y2r[t][j] + sigma * (xbv - zs[r][n]);
                y2r[t][j] = pv;
                sums[j] += pv * pv;
            }
        }
        #pragma unroll
        for (int j = 0; j < 8; ++j) atomicAdd(&pnorm[j + 8 * lh], sums[j]);

        // xbs columns are wave-private; same-wave LDS is in-order (DScnt).
        __builtin_amdgcn_wave_barrier();

        // ---- GEMM2: y1acc += xb @ S^T (K = own 64-column slab)
        v2f a2[16];
        #pragma unroll
        for (int ks = 0; ks < 16; ++ks) {
            a2[ks].x = xbs[l16][c0 + 4 * ks + 2 * lh];
            a2[ks].y = xbs[l16][c0 + 4 * ks + 2 * lh + 1];
        }
        #pragma unroll
        for (int t = 0; t < 4; ++t) {                         // N-tiles over m = 64
            v8f acc = {0.f,0.f,0.f,0.f,0.f,0.f,0.f,0.f};
            const int nn = t * 16 + l16;
            const uint4 Wa = *(const uint4*)&Ssp[nn][lh][kb];
            const uint4 Wb = *(const uint4*)&Ssp[nn][lh][kb + 8];
            const unsigned int wbuf[8] = {Wa.x, Wa.y, Wa.z, Wa.w,
                                          Wb.x, Wb.y, Wb.z, Wb.w};
            #pragma unroll
            for (int ks = 0; ks < 16; ++ks) {
                const unsigned int w = wbuf[ks >> 1];
                v2f b;
                if ((ks & 1) == 0) { b.x = (float)(w & 0xffu);
                                     b.y = (float)((w >> 8) & 0xffu); }
                else               { b.x = (float)((w >> 16) & 0xffu);
                                     b.y = (float)(w >> 24); }
                acc = __builtin_amdgcn_wmma_f32_16x16x4_f32(
                        false, a2[ks], false, b, (short)0, acc, false, false);
            }
            #pragma unroll
            for (int j = 0; j < 8; ++j)
                atomicAdd(&y1acc[j + 8 * lh][nn], acc[j]);
        }
        __syncthreads();   // y1acc + pnorm complete, xb reads done

        // ---- dual y1 update (4 fixed elements per thread, b in regs)
        #pragma unroll
        for (int q = 0; q < 4; ++q) {
            const int idx = tid + q * 256;
            const int r = idx >> 6, c = idx & 63;
            y1s[r][c] = fmaxf(y1s[r][c] + sigma * (y1acc[r][c] - br[q]), 0.0f);
        }
        // ---- dual y2 ball projection (registers)
        float sc[8];
        #pragma unroll
        for (int j = 0; j < 8; ++j) {
            const float pn = sqrtf(pnorm[j + 8 * lh]);
            sc[j] = fminf(1.0f, CONTROL / (pn + 1e-12f));
        }
        #pragma unroll
        for (int t = 0; t < 4; ++t)
            #pragma unroll
            for (int j = 0; j < 8; ++j)
                y2r[t][j] *= sc[j];
        __syncthreads();   // y1s stable before next iteration
    }

    // ---- write x (registers) to output ----
    #pragma unroll
    for (int t = 0; t < 4; ++t) {
        const int n = c0 + t * 16 + l16;
        #pragma unroll
        for (int j = 0; j < 8; ++j)
            out[(r0 + j + 8 * lh) * 512 + n] = xr[t][j];
    }
}

// ---------------------------------------------------------------------------
extern "C" void kernel_launch(void* const* d_in, const int* in_sizes, int n_in,
                              void* d_out, int out_size, void* d_ws, size_t ws_size,
                              hipStream_t stream)
{
    const float* X  = (const float*)d_in[0];
    const float* W1 = (const float*)d_in[1];
    const float* b1 = (const float*)d_in[2];
    const float* W2 = (const float*)d_in[3];
    const float* b2 = (const float*)d_in[4];
    const float* W3 = (const float*)d_in[5];
    const float* b3 = (const float*)d_in[6];
    const float* W4 = (const float*)d_in[7];
    const float* b4 = (const float*)d_in[8];
    const float* S  = (const float*)d_in[9];

    float* wsf   = (float*)d_ws;
    float* stepp = wsf;            // [1]
    float* z     = wsf + 256;      // [64*512]

    mlp_kernel<<<64, 128, 0, stream>>>(X, W1, b1, W2, b2, W3, b3, W4, b4, z);
    power_kernel<<<1, 512, 0, stream>>>(S, stepp);
    solver_kernel<<<4, 256, 0, stream>>>(X, S, z, stepp, (float*)d_out);
}